// GraphEncoder_8564164788771
// MI455X (gfx1250) — compile-verified
//
#include <hip/hip_runtime.h>

// GINE graph encoder for MI455X (gfx1250), wave32, f32 WMMA 16x16x4.
// N=50000 nodes, E=800000 edges, 64 features, 3 layers.

#define NN 50000
#define EE 800000
#define F  64
#define LAYERS 3
#define LN_EPS 1e-5f

typedef __attribute__((ext_vector_type(2))) float v2f;
typedef __attribute__((ext_vector_type(8))) float v8f;

__device__ __forceinline__ v8f wmma_f32_16x16x4(v2f a, v2f b, v8f c) {
  // D = A(16x4) * B(4x16) + C(16x16), all f32, wave32
  return __builtin_amdgcn_wmma_f32_16x16x4_f32(
      /*neg_a=*/false, a, /*neg_b=*/false, b,
      /*c_mod=*/(short)0, c, /*reuse_a=*/false, /*reuse_b=*/false);
}

// ---------------------------------------------------------------------------
// Fuse edge encoder into per-layer edge-linear:
//   W'[l] = enc_W (4x64) @ edge_lin_W[l] (64x64)   -> [L,4,64]
//   b'[l] = enc_b @ edge_lin_W[l] + edge_lin_b[l]  -> [L,64]
// ---------------------------------------------------------------------------
__global__ void __launch_bounds__(256)
fuse_weights_kernel(const float* __restrict__ enc_W, const float* __restrict__ enc_b,
                    const float* __restrict__ elW, const float* __restrict__ elb,
                    float* __restrict__ Wp, float* __restrict__ bp) {
  int t = blockIdx.x * blockDim.x + threadIdx.x;
  if (t >= LAYERS * F) return;
  int l = t >> 6;          // layer
  int n = t & 63;          // output column
  const float* Wl = elW + l * F * F;
  #pragma unroll
  for (int k = 0; k < 4; ++k) {
    float acc = 0.f;
    for (int j = 0; j < F; ++j) acc += enc_W[k * F + j] * Wl[j * F + n];
    Wp[l * 4 * F + k * F + n] = acc;
  }
  float bacc = elb[l * F + n];
  for (int j = 0; j < F; ++j) bacc += enc_b[j] * Wl[j * F + n];
  bp[l * F + n] = bacc;
}

__global__ void fill_zero_kernel(float* __restrict__ p, int n) {
  int i = blockIdx.x * blockDim.x + threadIdx.x;
  int stride = gridDim.x * blockDim.x;
  for (; i < n; i += stride) p[i] = 0.f;
}

// ---------------------------------------------------------------------------
// Edge kernel: one wave per 16-edge tile.
//   e = edge_attr(16x4) @ W'(4x64) + b'   -> one K=4 WMMA per 16-col tile
//   msg = relu(h[src] + e); atomicAdd into agg[dst]
// E = 800000 = 50000 tiles exactly; 8 waves/block -> 6250 blocks.
// ---------------------------------------------------------------------------
__global__ void __launch_bounds__(256)
gine_edge_kernel(const float* __restrict__ edge_attr,
                 const int* __restrict__ edge_index,
                 const float* __restrict__ Wp,      // this layer's fused W' [4,64]
                 const float* __restrict__ bp,      // this layer's fused b' [64]
                 const float* __restrict__ h_in,    // [N,64]
                 float* __restrict__ agg,           // [N,64]
                 int ntiles) {
  int wave = (blockIdx.x * blockDim.x + threadIdx.x) >> 5;
  int lane = threadIdx.x & 31;
  if (wave >= ntiles) return;                       // uniform per wave, no barriers here
  int edge0 = wave * 16;
  int half = lane >> 4;                             // 0: K0/K1, 1: K2/K3
  int l16  = lane & 15;

  // A operand: 16x4 tile of edge_attr (row = l16, K pair by half)
  v2f a;
  {
    const float* ap = edge_attr + (size_t)(edge0 + l16) * 4 + 2 * half;
    a.x = ap[0]; a.y = ap[1];
  }

  v8f acc[4];
  #pragma unroll
  for (int nt = 0; nt < 4; ++nt) {
    int col = nt * 16 + l16;
    v2f b;
    b.x = Wp[(2 * half + 0) * F + col];
    b.y = Wp[(2 * half + 1) * F + col];
    v8f c = {};
    acc[nt] = wmma_f32_16x16x4(a, b, c);
  }

  // D layout: lane holds col = nt*16 + l16; vgpr r holds row m = r + 8*half
  #pragma unroll
  for (int r = 0; r < 8; ++r) {
    int m  = r + 8 * half;
    int em = edge0 + m;
    int s  = edge_index[em];        // src
    int d  = edge_index[EE + em];   // dst
    #pragma unroll
    for (int nt = 0; nt < 4; ++nt) {
      int col = nt * 16 + l16;
      float msg = h_in[(size_t)s * F + col] + acc[nt][r] + bp[col];
      msg = msg > 0.f ? msg : 0.f;
      atomicAdd(&agg[(size_t)d * F + col], msg);
    }
  }
}

// ---------------------------------------------------------------------------
// Node kernel: one wave per 16-node tile.
//   z = (1+eps)h + agg           (staged in LDS, stride 65 = conflict-free)
//   t = relu(z @ W1 + b1)        (16 K-steps of WMMA f32 16x16x4)
//   u = relu(t @ W2 + b2) + h
//   h' = LayerNorm(u) * g + b    (half-wave shfl_xor reductions)
// N = 50000 = 3125 tiles; 8 waves/block -> 391 blocks; tail waves clamped.
// ---------------------------------------------------------------------------
#define ZSTRIDE 65   // 64 + 1 padding float per row -> distinct LDS banks

__global__ void __launch_bounds__(256)
gine_node_kernel(const float* __restrict__ h_in, const float* __restrict__ agg,
                 const float* __restrict__ eps_arr, int layer,
                 const float* __restrict__ W1, const float* __restrict__ b1,
                 const float* __restrict__ W2, const float* __restrict__ b2,
                 const float* __restrict__ ln_g, const float* __restrict__ ln_b,
                 float* __restrict__ h_out, int ntiles) {
  __shared__ float smem[8][16 * ZSTRIDE];
  int wslot = threadIdx.x >> 5;
  int lane  = threadIdx.x & 31;
  int wave  = (blockIdx.x * blockDim.x + threadIdx.x) >> 5;
  if (wave >= ntiles) wave = ntiles - 1;   // redundant recompute keeps EXEC/barriers legal
  int node0 = wave * 16;
  int half = lane >> 4;
  int l16  = lane & 15;
  float* Z = smem[wslot];

  float epsl = 1.0f + eps_arr[layer];
  const float* W1l = W1 + (size_t)layer * F * F;
  const float* W2l = W2 + (size_t)layer * F * F;
  const float* b1l = b1 + layer * F;
  const float* b2l = b2 + layer * F;
  const float* gl  = ln_g + layer * F;
  const float* bl  = ln_b + layer * F;

  // stage z = (1+eps)*h + agg into LDS (coalesced global, padded LDS rows)
  for (int i = lane; i < 16 * F; i += 32) {
    int m = i >> 6, c = i & 63;
    Z[m * ZSTRIDE + c] = epsl * h_in[(size_t)node0 * F + i] + agg[(size_t)node0 * F + i];
  }
  __syncthreads();

  // MLP layer 1: acc = z @ W1
  v8f acc[4];
  #pragma unroll
  for (int nt = 0; nt < 4; ++nt) { v8f c = {}; acc[nt] = c; }
  for (int kk = 0; kk < 16; ++kk) {
    int k0 = 4 * kk + 2 * half;
    v2f a;
    a.x = Z[l16 * ZSTRIDE + k0];
    a.y = Z[l16 * ZSTRIDE + k0 + 1];
    #pragma unroll
    for (int nt = 0; nt < 4; ++nt) {
      int col = nt * 16 + l16;
      v2f b;
      b.x = W1l[(k0 + 0) * F + col];
      b.y = W1l[(k0 + 1) * F + col];
      acc[nt] = wmma_f32_16x16x4(a, b, acc[nt]);
    }
  }
  __syncthreads();   // all reads of z done before overwriting with t

  // t = relu(acc + b1) back into LDS (re-materialize A layout for MLP2)
  #pragma unroll
  for (int r = 0; r < 8; ++r) {
    int m = r + 8 * half;
    #pragma unroll
    for (int nt = 0; nt < 4; ++nt) {
      int col = nt * 16 + l16;
      float t = acc[nt][r] + b1l[col];
      Z[m * ZSTRIDE + col] = t > 0.f ? t : 0.f;
    }
  }
  __syncthreads();

  // MLP layer 2: acc2 = t @ W2
  v8f acc2[4];
  #pragma unroll
  for (int nt = 0; nt < 4; ++nt) { v8f c = {}; acc2[nt] = c; }
  for (int kk = 0; kk < 16; ++kk) {
    int k0 = 4 * kk + 2 * half;
    v2f a;
    a.x = Z[l16 * ZSTRIDE + k0];
    a.y = Z[l16 * ZSTRIDE + k0 + 1];
    #pragma unroll
    for (int nt = 0; nt < 4; ++nt) {
      int col = nt * 16 + l16;
      v2f b;
      b.x = W2l[(k0 + 0) * F + col];
      b.y = W2l[(k0 + 1) * F + col];
      acc2[nt] = wmma_f32_16x16x4(a, b, acc2[nt]);
    }
  }

  // u = relu(acc2 + b2) + h ; accumulate row sums for LayerNorm
  float sum[8], sumsq[8];
  #pragma unroll
  for (int r = 0; r < 8; ++r) { sum[r] = 0.f; sumsq[r] = 0.f; }
  #pragma unroll
  for (int r = 0; r < 8; ++r) {
    int m = r + 8 * half;
    #pragma unroll
    for (int nt = 0; nt < 4; ++nt) {
      int col = nt * 16 + l16;
      float v = acc2[nt][r] + b2l[col];
      v = v > 0.f ? v : 0.f;
      v += h_in[(size_t)(node0 + m) * F + col];
      acc2[nt][r] = v;
      sum[r] += v;
      sumsq[r] += v * v;
    }
  }
  // reduce across the 16 lanes holding each row (xor masks stay within half-wave)
  #pragma unroll
  for (int r = 0; r < 8; ++r) {
    #pragma unroll
    for (int off = 1; off < 16; off <<= 1) {
      sum[r]   += __shfl_xor(sum[r],   off, 32);
      sumsq[r] += __shfl_xor(sumsq[r], off, 32);
    }
  }
  const float inv64 = 1.0f / 64.0f;
  #pragma unroll
  for (int r = 0; r < 8; ++r) {
    float mean = sum[r] * inv64;
    float var  = sumsq[r] * inv64 - mean * mean;
    float inv  = rsqrtf(var + LN_EPS);
    int m = r + 8 * half;
    #pragma unroll
    for (int nt = 0; nt < 4; ++nt) {
      int col = nt * 16 + l16;
      h_out[(size_t)(node0 + m) * F + col] =
          (acc2[nt][r] - mean) * inv * gl[col] + bl[col];
    }
  }
}

// ---------------------------------------------------------------------------
extern "C" void kernel_launch(void* const* d_in, const int* in_sizes, int n_in,
                              void* d_out, int out_size, void* d_ws, size_t ws_size,
                              hipStream_t stream) {
  (void)in_sizes; (void)n_in; (void)out_size; (void)ws_size;

  const float* x          = (const float*)d_in[0];
  const int*   edge_index = (const int*)  d_in[1];
  const float* edge_attr  = (const float*)d_in[2];
  const float* enc_W      = (const float*)d_in[3];
  const float* enc_b      = (const float*)d_in[4];
  const float* eps        = (const float*)d_in[5];
  const float* elW        = (const float*)d_in[6];
  const float* elb        = (const float*)d_in[7];
  const float* W1         = (const float*)d_in[8];
  const float* b1         = (const float*)d_in[9];
  const float* W2         = (const float*)d_in[10];
  const float* b2         = (const float*)d_in[11];
  const float* lng        = (const float*)d_in[12];
  const float* lnb        = (const float*)d_in[13];
  float* out = (float*)d_out;

  // workspace layout (floats): W'[3*4*64] | b'[3*64] | agg[N*64] | hA[N*64] | hB[N*64]
  float* ws  = (float*)d_ws;
  float* Wp  = ws;
  float* bp  = Wp + LAYERS * 4 * F;
  float* agg = bp + LAYERS * F;
  float* hA  = agg + (size_t)NN * F;
  float* hB  = hA  + (size_t)NN * F;

  fuse_weights_kernel<<<1, 256, 0, stream>>>(enc_W, enc_b, elW, elb, Wp, bp);

  const int edge_tiles = EE / 16;                 // 50000
  const int node_tiles = NN / 16 + (NN % 16 ? 1 : 0);  // 3125
  const int edge_blocks = edge_tiles / 8;         // 6250 (exact)
  const int node_blocks = (node_tiles + 7) / 8;   // 391

  const float* hin = x;
  float* houts[LAYERS] = { hA, hB, out };
  for (int l = 0; l < LAYERS; ++l) {
    fill_zero_kernel<<<512, 256, 0, stream>>>(agg, NN * F);
    gine_edge_kernel<<<edge_blocks, 256, 0, stream>>>(
        edge_attr, edge_index, Wp + l * 4 * F, bp + l * F, hin, agg, edge_tiles);
    gine_node_kernel<<<node_blocks, 256, 0, stream>>>(
        hin, agg, eps, l, W1, b1, W2, b2, lng, lnb, houts[l], node_tiles);
    hin = houts[l];
  }
}